// SASA_Layer_71665824301507
// MI455X (gfx1250) — compile-verified
//
#include <hip/hip_runtime.h>
#include <math.h>

typedef float v2f __attribute__((ext_vector_type(2)));
typedef float v8f __attribute__((ext_vector_type(8)));

#define CCH  128                 // channels
#define SDIM 24                  // spatial extent
#define NVOX (SDIM * SDIM * SDIM) // 13824 voxels

// ---------------------------------------------------------------------------
// Stage 0: precompute the 27 window-bias scalars
//   bias27[s] = sum_c rel_h[c,i] + sum_c rel_w[c,j] + sum_c rel_d[c,l]
//   s = (i*3 + j)*3 + l,  rel_* are (64,3) row-major
// ---------------------------------------------------------------------------
__global__ void bias27_kernel(const float* __restrict__ rh,
                              const float* __restrict__ rw,
                              const float* __restrict__ rd,
                              float* __restrict__ bias27) {
    int s = threadIdx.x;
    if (s >= 27) return;
    int i = s / 9, j = (s % 9) / 3, l = s % 3;
    float acc = 0.0f;
    for (int c = 0; c < 64; ++c)
        acc += rh[c * 3 + i] + rw[c * 3 + j] + rd[c * 3 + l];
    bias27[s] = acc;
}

// ---------------------------------------------------------------------------
// Stage 1: 1x1x1 conv == dense GEMM  Y[n,o] = sum_c W[o,c]*X[c,n] + b[o]
// via V_WMMA_F32_16X16X4_F32 (exact fp32 matrix pipe).
// Block = 256 threads = 8 waves; wave w owns M-tile (out-channels) 16w,
// blockIdx.x owns N-tile (voxels) of 16. blockIdx.y selects q/k/v weights.
// K loop: 128 in chunks of 4 -> 32 WMMAs per tile.
//
// fp32 A 16x4 layout: lanes 0-15 M=lane, a0->K=2*half+0, a1->K=2*half+1
// fp32 B 4x16 layout: lanes 0-15 N=lane, b0->K=2*half+0, b1->K=2*half+1
// fp32 C/D 16x16:     VGPR r, lane half: M = r + 8*half, N = lane%16
// ---------------------------------------------------------------------------
__global__ __launch_bounds__(256) void conv1x1_wmma(
    const float* __restrict__ X,                               // [C, NVOX]
    const float* __restrict__ Wq, const float* __restrict__ Bq,
    const float* __restrict__ Wk, const float* __restrict__ Bk,
    const float* __restrict__ Wv, const float* __restrict__ Bv,
    float* __restrict__ Yq, float* __restrict__ Yk, float* __restrict__ Yv) {
    const float* W; const float* Bb; float* Y;
    if (blockIdx.y == 0)      { W = Wq; Bb = Bq; Y = Yq; }
    else if (blockIdx.y == 1) { W = Wk; Bb = Bk; Y = Yk; }
    else                      { W = Wv; Bb = Bv; Y = Yv; }

    const int wave = threadIdx.x >> 5;
    const int lane = threadIdx.x & 31;
    const int half = lane >> 4;   // 0: K pair {0,1}, 1: K pair {2,3}
    const int lm   = lane & 15;
    const int m0   = wave * 16;           // out-channel tile
    const int n0   = blockIdx.x * 16;     // voxel tile

    // seed accumulator with bias (D = A*B + C)
    v8f acc;
#pragma unroll
    for (int r = 0; r < 8; ++r) acc[r] = Bb[m0 + r + 8 * half];

    const float* wrow = W + (size_t)(m0 + lm) * CCH;   // row of W for M=lm
    const float* xcol = X + (n0 + lm);                 // column of X for N=lm

#pragma unroll 4
    for (int k0 = 0; k0 < CCH; k0 += 4) {
        v2f a, b;
        a[0] = wrow[k0 + 2 * half + 0];
        a[1] = wrow[k0 + 2 * half + 1];
        b[0] = xcol[(size_t)(k0 + 2 * half + 0) * NVOX];
        b[1] = xcol[(size_t)(k0 + 2 * half + 1) * NVOX];
        acc = __builtin_amdgcn_wmma_f32_16x16x4_f32(
            /*neg_a=*/false, a, /*neg_b=*/false, b,
            /*c_mod=*/(short)0, acc, /*reuse_a=*/false, /*reuse_b=*/false);
    }

    // store voxel-major, channel-contiguous: Y[(n0+lm)*128 + m]
    float* yrow = Y + (size_t)(n0 + lm) * CCH + m0 + 8 * half;
#pragma unroll
    for (int r = 0; r < 8; ++r) yrow[r] = acc[r];
}

// ---------------------------------------------------------------------------
// Stage 2: windowed attention. One wave32 per voxel, 8 voxels per block.
// Lane l owns channels {l, l+32, l+64, l+96}. Out-of-range neighbors use
// the conv bias vectors bk/bv (conv of zero padding). Output is written in
// [voxel, channel] order, which is exactly the reference's reshape layout.
// ---------------------------------------------------------------------------
__global__ __launch_bounds__(256) void attn_kernel(
    const float* __restrict__ q,  const float* __restrict__ kf,
    const float* __restrict__ vf, const float* __restrict__ bk,
    const float* __restrict__ bv, const float* __restrict__ bias27,
    float* __restrict__ out) {
    const int wave = threadIdx.x >> 5;
    const int lane = threadIdx.x & 31;
    const int vox  = blockIdx.x * 8 + wave;
    const int h = vox / (SDIM * SDIM);
    const int w = (vox / SDIM) % SDIM;
    const int d = vox % SDIM;

    float qv[4];
#pragma unroll
    for (int k = 0; k < 4; ++k) qv[k] = q[(size_t)vox * CCH + lane + 32 * k];

    // qsum over all 128 channels (butterfly across the wave)
    float qs = qv[0] + qv[1] + qv[2] + qv[3];
#pragma unroll
    for (int m = 16; m >= 1; m >>= 1) qs += __shfl_xor(qs, m, 32);

    // qk[s] = q . k_neighbor + qsum * bias27[s]; lane s ends up holding qk[s]
    float myqk = 0.0f;
    for (int s = 0; s < 27; ++s) {
        int nh = h + s / 9 - 1, nw = w + (s % 9) / 3 - 1, nd = d + s % 3 - 1;
        bool in = (unsigned)nh < SDIM && (unsigned)nw < SDIM && (unsigned)nd < SDIM;
        const float* kp = in ? (kf + (size_t)((nh * SDIM + nw) * SDIM + nd) * CCH) : bk;
        float p = 0.0f;
#pragma unroll
        for (int k = 0; k < 4; ++k) p += qv[k] * kp[lane + 32 * k];
#pragma unroll
        for (int m = 16; m >= 1; m >>= 1) p += __shfl_xor(p, m, 32);
        if (lane == s) myqk = p + qs * bias27[s];
    }

    // softmax over the 27 taps, held one-per-lane
    float val = (lane < 27) ? myqk : -INFINITY;
    float mx = val;
#pragma unroll
    for (int m = 16; m >= 1; m >>= 1) mx = fmaxf(mx, __shfl_xor(mx, m, 32));
    float e = (lane < 27) ? __expf(val - mx) : 0.0f;
    float sum = e;
#pragma unroll
    for (int m = 16; m >= 1; m >>= 1) sum += __shfl_xor(sum, m, 32);
    float attn = e / sum;

    // out[c] = sum_s attn[s] * v_neighbor[c]
    float acc[4] = {0.f, 0.f, 0.f, 0.f};
    for (int s = 0; s < 27; ++s) {
        float a = __shfl(attn, s, 32);
        int nh = h + s / 9 - 1, nw = w + (s % 9) / 3 - 1, nd = d + s % 3 - 1;
        bool in = (unsigned)nh < SDIM && (unsigned)nw < SDIM && (unsigned)nd < SDIM;
        const float* vp = in ? (vf + (size_t)((nh * SDIM + nw) * SDIM + nd) * CCH) : bv;
#pragma unroll
        for (int k = 0; k < 4; ++k) acc[k] += a * vp[lane + 32 * k];
    }
#pragma unroll
    for (int k = 0; k < 4; ++k) out[(size_t)vox * CCH + lane + 32 * k] = acc[k];
}

// ---------------------------------------------------------------------------
extern "C" void kernel_launch(void* const* d_in, const int* in_sizes, int n_in,
                              void* d_out, int out_size, void* d_ws, size_t ws_size,
                              hipStream_t stream) {
    const float* x  = (const float*)d_in[0];
    const float* wq = (const float*)d_in[1];
    const float* bq = (const float*)d_in[2];
    const float* wk = (const float*)d_in[3];
    const float* bk = (const float*)d_in[4];
    const float* wv = (const float*)d_in[5];
    const float* bv = (const float*)d_in[6];
    const float* rh = (const float*)d_in[7];
    const float* rw = (const float*)d_in[8];
    const float* rd = (const float*)d_in[9];
    float* out = (float*)d_out;

    float* q      = (float*)d_ws;                    // [NVOX, C]
    float* kf     = q  + (size_t)NVOX * CCH;         // [NVOX, C]
    float* vf     = kf + (size_t)NVOX * CCH;         // [NVOX, C]
    float* bias27 = vf + (size_t)NVOX * CCH;         // [27]

    bias27_kernel<<<1, 32, 0, stream>>>(rh, rw, rd, bias27);
    conv1x1_wmma<<<dim3(NVOX / 16, 3), 256, 0, stream>>>(
        x, wq, bq, wk, bk, wv, bv, q, kf, vf);
    attn_kernel<<<NVOX / 8, 256, 0, stream>>>(q, kf, vf, bk, bv, bias27, out);
}